// RegionProposalLoss_80788334838310
// MI455X (gfx1250) — compile-verified
//
#include <hip/hip_runtime.h>
#include <stdint.h>

typedef __attribute__((ext_vector_type(2))) float v2f;
typedef __attribute__((ext_vector_type(8))) float v8f;

#define HW_SHIFT 20            // H*W = 1024*1024 pixels per batch image
#define ACC_STRIDE 32          // per-batch counter stride (25 ids padded to 32)
#define ACC_BYTES 4096         // reserved counter area at start of workspace
#define MAX_B 16

__device__ __forceinline__ uint64_t mix64(uint64_t h) {
  h ^= h >> 33; h *= 0xff51afd7ed558ccdULL;
  h ^= h >> 33; h *= 0xc4ceb9fe1a85ec53ULL;
  h ^= h >> 33;
  return h;
}

// Pass 1: hash-set dedup of (batch, id, value_bits); per-block LDS counters,
// flushed once per block to the global per-(b,id) unique-count table.
__global__ void region_unique_count_kernel(
    const float* __restrict__ val, const int* __restrict__ ids,
    unsigned long long* __restrict__ table, unsigned table_mask,
    unsigned* __restrict__ gcount, int npix)
{
  __shared__ unsigned lcnt[MAX_B * ACC_STRIDE];
  for (int j = threadIdx.x; j < MAX_B * ACC_STRIDE; j += blockDim.x) lcnt[j] = 0u;
  __syncthreads();

  const int stride = gridDim.x * blockDim.x * 4;
  for (int i = (blockIdx.x * blockDim.x + threadIdx.x) * 4; i < npix; i += stride) {
    // 16B vector loads; a group of 4 never straddles a batch boundary (HW % 4 == 0)
    const int4  id4 = *(const int4*)(ids + i);
    const uint4 v4  = *(const uint4*)(val + i);
    const unsigned b = (unsigned)(i >> HW_SHIFT);
#pragma unroll
    for (int p = 0; p < 4; ++p) {
      const int id = (&id4.x)[p];
      if (id == 0) continue;                       // id 0 excluded from the loss
      unsigned bits = (&v4.x)[p];
      if (bits == 0x80000000u) bits = 0u;          // -0.0f compares equal to +0.0f
      const unsigned seg = (b << 5) | (unsigned)id;
      const uint64_t key = ((uint64_t)seg << 32) | (uint64_t)bits; // nonzero (id>=1)
      unsigned slot = (unsigned)mix64(key) & table_mask;
      for (;;) {
        unsigned long long old =
            atomicCAS(&table[slot], 0ull, (unsigned long long)key);
        if (old == 0ull) { atomicAdd(&lcnt[seg], 1u); break; }  // new distinct value
        if (old == (unsigned long long)key) break;              // duplicate
        slot = (slot + 1u) & table_mask;                        // linear probe
      }
    }
  }

  __syncthreads();
  for (int j = threadIdx.x; j < MAX_B * ACC_STRIDE; j += blockDim.x) {
    const unsigned c = lcnt[j];
    if (c) atomicAdd(&gcount[j], c);
  }
}

// Pass 2 (one wave32): WMMA row-sum reduction over the 16x24 per-(b,id) stats,
// then the serial scan loss = ((L0)/n0 + L1)/n1 + ... matching jax.lax.scan.
__global__ void region_finalize_kernel(const unsigned* __restrict__ gcount,
                                       float* __restrict__ out)
{
  const int lane = (int)threadIdx.x;   // 0..31, EXEC all-ones through the WMMAs
  const int b    = lane & 15;          // M (batch) index per A-layout
  const int half = lane >> 4;          // lanes 0-15: K=0,1 ; lanes 16-31: K=2,3

  v2f onesB; onesB.x = 1.0f; onesB.y = 1.0f;   // all-ones B => D[m][n] = rowsum(A[m])
  v8f accL = {};                                // sum of (nuniq - 1) per batch
  v8f accN = {};                                // count of present ids per batch

#pragma unroll
  for (int c = 0; c < 6; ++c) {                 // 6 chunks of K=4 cover ids 1..24
    float aL[2], aN[2];
#pragma unroll
    for (int q = 0; q < 2; ++q) {
      const int k  = c * 4 + half * 2 + q;      // 0..23
      const int id = k + 1;                     // 1..24
      const unsigned u = gcount[(b << 5) | id];
      aL[q] = (u > 0u) ? (float)(u - 1u) : 0.0f;
      aN[q] = (u > 0u) ? 1.0f : 0.0f;
    }
    v2f AL; AL.x = aL[0]; AL.y = aL[1];
    v2f AN; AN.x = aN[0]; AN.y = aN[1];
    accL = __builtin_amdgcn_wmma_f32_16x16x4_f32(
        false, AL, false, onesB, (short)0, accL, false, false);
    accN = __builtin_amdgcn_wmma_f32_16x16x4_f32(
        false, AN, false, onesB, (short)0, accN, false, false);
  }

  // D layout: VGPR r, lanes 0-15 -> M=r ; lanes 16-31 -> M=r+8. Take N=0 column.
  __shared__ float sL[16], sN[16];
  if (lane == 0) {
#pragma unroll
    for (int r = 0; r < 8; ++r) { sL[r] = accL[r]; sN[r] = accN[r]; }
  }
  if (lane == 16) {
#pragma unroll
    for (int r = 0; r < 8; ++r) { sL[r + 8] = accL[r]; sN[r + 8] = accN[r]; }
  }
  __syncthreads();

  if (lane == 0) {
    float loss = 0.0f;
#pragma unroll
    for (int bb = 0; bb < 16; ++bb) {
      loss += sL[bb];
      if (sN[bb] > 0.0f) loss /= sN[bb];
    }
    out[0] = loss;
  }
}

extern "C" void kernel_launch(void* const* d_in, const int* in_sizes, int n_in,
                              void* d_out, int out_size, void* d_ws, size_t ws_size,
                              hipStream_t stream) {
  // setup_inputs order: tgt_img(f32), y_tgt_pr(f32), mask(i32), weak_mask(i32)
  const float* ypr   = (const float*)d_in[1];
  const int*   wmask = (const int*)d_in[3];
  const int    npix  = in_sizes[3];            // B * H * W (16 * 1024 * 1024)

  unsigned*           gcount = (unsigned*)d_ws;
  unsigned long long* table  =
      (unsigned long long*)((char*)d_ws + ACC_BYTES);

  // Hash table sized from workspace: power-of-two slots, target 2^25 (256MB,
  // load factor ~0.5 for 16M keys, mostly L2-resident in the 192MB L2).
  const size_t avail = (ws_size > (size_t)ACC_BYTES)
                           ? (ws_size - (size_t)ACC_BYTES) / 8u : 1u;
  unsigned long long slots = 1ull;
  while ((slots << 1) <= (unsigned long long)avail && slots < (1ull << 25))
    slots <<= 1;
  const unsigned tmask = (unsigned)(slots - 1ull);

  hipMemsetAsync(d_ws, 0, (size_t)ACC_BYTES + (size_t)slots * 8u, stream);

  dim3 block(256);
  dim3 grid(2048);   // grid-stride, 4 pixels/thread/iter -> ~8 passes over 16M
  hipLaunchKernelGGL(region_unique_count_kernel, grid, block, 0, stream,
                     ypr, wmask, table, tmask, gcount, npix);
  hipLaunchKernelGGL(region_finalize_kernel, dim3(1), dim3(32), 0, stream,
                     gcount, (float*)d_out);
}